// DeformableConvNet_20675972563616
// MI455X (gfx1250) — compile-verified
//
#include <hip/hip_runtime.h>
#include <hip/hip_bf16.h>

// ---------------------------------------------------------------------------
// Deformable conv (B=8, C=128, H=W=64, O=128, K=3, pad=1) for gfx1250.
// Stage 2 GEMM (M=128, N=32768, K=1152) with V_WMMA_F32_16X16X4_F32, fused
// with bilinear sampling through LDS.  N tile widened to 32 pixels so each
// weight (A) fragment feeds two WMMA accumulators -> halves L2 weight traffic.
// ---------------------------------------------------------------------------

#define BATCH 8
#define CIN   128
#define HDIM  64
#define WDIM  64
#define HW    4096          // HDIM*WDIM
#define OCH   128
#define K2    9
#define KC    1152          // K2*CIN, GEMM reduction depth
#define NPIX  32            // pixels per workgroup (two 16-wide N sub-tiles)
#define SPAD  33            // LDS row stride (floats) for the sampled tile

typedef float v2f __attribute__((ext_vector_type(2)));
typedef float v8f __attribute__((ext_vector_type(8)));

// ---------------------------------------------------------------------------
// Kernel 1: x NCHW -> NHWC  (makes bilinear corner fetches 512B contiguous)
// ---------------------------------------------------------------------------
__global__ void xpose_x_kernel(const float* __restrict__ x, float* __restrict__ xn) {
    int idx = blockIdx.x * 256 + threadIdx.x;      // over B*C*HW = 4194304
    int hw  = idx & (HW - 1);
    int bc  = idx >> 12;
    int c   = bc & (CIN - 1);
    int b   = bc >> 7;
    xn[(size_t)(b * HW + hw) * CIN + c] = x[idx];  // coalesced read, strided write
}

// ---------------------------------------------------------------------------
// Kernel 2: weight (O,C,3,3) -> w2[o][k*128+c]; offset_w (18,C,3,3) -> ow2.
// Linearizes the GEMM reduction index kc = k*128 + c.
// ---------------------------------------------------------------------------
__global__ void xpose_w_kernel(const float* __restrict__ w,  float* __restrict__ w2,
                               const float* __restrict__ ow, float* __restrict__ ow2) {
    int idx = blockIdx.x * 256 + threadIdx.x;
    if (idx < OCH * KC) {                 // 147456
        int k  = idx % K2;
        int oc = idx / K2;
        int c  = oc & (CIN - 1);
        int o  = oc >> 7;
        w2[(size_t)o * KC + k * CIN + c] = w[idx];
    }
    if (idx < 18 * KC) {                  // 20736
        int k  = idx % K2;
        int jc = idx / K2;
        int c  = jc & (CIN - 1);
        int j  = jc >> 7;
        ow2[(size_t)j * KC + k * CIN + c] = ow[idx];
    }
}

// ---------------------------------------------------------------------------
// Kernel 3: offset conv.  One thread = one pixel, 18 accumulators.
// Weight addresses are wave-uniform -> scalar-cache friendly.
// ---------------------------------------------------------------------------
__global__ void __launch_bounds__(256)
offset_conv_kernel(const float* __restrict__ xn, const float* __restrict__ ow2,
                   const float* __restrict__ ob, float* __restrict__ offs) {
    int g  = blockIdx.x * 256 + threadIdx.x;  // pixel over B*HW = 32768
    int b  = g >> 12;
    int hw = g & (HW - 1);
    int h  = hw >> 6;
    int w  = hw & 63;

    float acc[18];
    #pragma unroll
    for (int j = 0; j < 18; ++j) acc[j] = ob[j];

    for (int k = 0; k < K2; ++k) {
        int kh = k / 3, kw = k % 3;
        int y = h - 1 + kh, x = w - 1 + kw;
        if ((unsigned)y < 64u && (unsigned)x < 64u) {
            const float* xp = xn + (size_t)(b * HW + y * WDIM + x) * CIN;
            for (int c = 0; c < CIN; c += 4) {
                float4 xv = *(const float4*)(xp + c);
                #pragma unroll
                for (int j = 0; j < 18; ++j) {
                    const float* wp = ow2 + (size_t)j * KC + k * CIN + c;
                    acc[j] += xv.x * wp[0] + xv.y * wp[1] + xv.z * wp[2] + xv.w * wp[3];
                }
            }
        }
    }
    #pragma unroll
    for (int j = 0; j < 18; ++j)
        offs[(size_t)(b * 18 + j) * HW + hw] = acc[j];
}

// ---------------------------------------------------------------------------
// Kernel 4: fused bilinear sampling + WMMA GEMM.
// One workgroup = 32 pixels. 256 threads / 8 waves.
// Phase 1: build S[kc][pix] in LDS (1152 x 32, stride 33 floats = 148.5 KB),
//   two 16-pixel halves; 16 consecutive lanes share a pixel -> coalesced
//   512B corner vectors.
// Phase 2: wave m handles output rows [16m,16m+16) against BOTH 16-pixel
//   N sub-tiles with a single A fragment per step (A reuse x2).
//   f32 16x4 A layout (ISA 7.12.2): lane<16 -> K={0,1}, lane>=16 -> K={2,3}.
// ---------------------------------------------------------------------------
__global__ void __launch_bounds__(256)
deform_wmma_kernel(const float* __restrict__ xn, const float* __restrict__ offs,
                   const float* __restrict__ w2, const float* __restrict__ bias,
                   float* __restrict__ out) {
    __shared__ float S[KC * SPAD];   // 38016 floats = 148.5 KB (of 320 KB/WGP)

    const int tid  = threadIdx.x;
    const int tile = blockIdx.x;     // 1024 tiles of 32 pixels

    // ---- phase 1: sample into LDS ----
    {
        const int cg = tid & 15;     // channel group (8 channels)
        #pragma unroll
        for (int half = 0; half < 2; ++half) {
            const int p  = (tid >> 4) + half * 16;   // pixel in tile (0..31)
            const int g  = tile * NPIX + p;
            const int b  = g >> 12;
            const int hw = g & (HW - 1);
            const int h  = hw >> 6;
            const int w  = hw & 63;
            const float* xb = xn + (size_t)b * HW * CIN;

            for (int k = 0; k < K2; ++k) {
                float dy = offs[(size_t)(b * 18 + 2 * k + 0) * HW + hw];
                float dx = offs[(size_t)(b * 18 + 2 * k + 1) * HW + hw];
                float ys = (float)(h - 1 + (k / 3)) + dy;
                float xs = (float)(w - 1 + (k % 3)) + dx;
                float y0f = floorf(ys), x0f = floorf(xs);
                int   iy0 = (int)y0f,   ix0 = (int)x0f;
                float wy1 = ys - y0f,   wx1 = xs - x0f;
                float wy0 = 1.0f - wy1, wx0 = 1.0f - wx1;

                const float cw[4] = { wy0 * wx0, wy0 * wx1, wy1 * wx0, wy1 * wx1 };
                const int   iy[4] = { iy0, iy0, iy0 + 1, iy0 + 1 };
                const int   ix[4] = { ix0, ix0 + 1, ix0, ix0 + 1 };

                float v[8] = {0.f, 0.f, 0.f, 0.f, 0.f, 0.f, 0.f, 0.f};
                #pragma unroll
                for (int cnr = 0; cnr < 4; ++cnr) {
                    if ((unsigned)iy[cnr] < 64u && (unsigned)ix[cnr] < 64u) {
                        const float* cp = xb + (size_t)(iy[cnr] * WDIM + ix[cnr]) * CIN + cg * 8;
                        float4 a0 = *(const float4*)cp;
                        float4 a1 = *(const float4*)(cp + 4);
                        float  wt = cw[cnr];
                        v[0] += wt * a0.x; v[1] += wt * a0.y; v[2] += wt * a0.z; v[3] += wt * a0.w;
                        v[4] += wt * a1.x; v[5] += wt * a1.y; v[6] += wt * a1.z; v[7] += wt * a1.w;
                    }
                }
                const int rbase = k * CIN + cg * 8;
                #pragma unroll
                for (int i = 0; i < 8; ++i)
                    S[(rbase + i) * SPAD + p] = v[i];
            }
        }
    }
    __syncthreads();

    // ---- phase 2: WMMA GEMM, one 16-row M tile per wave, two N sub-tiles ----
    const int wave = tid >> 5;       // 0..7 -> output rows [16*wave, 16*wave+16)
    const int lane = tid & 31;
    const int nl   = lane & 15;      // M row (A) / N col (B,D) within tile
    const int hi   = lane >> 4;      // 0: K={0,1}, 1: K={2,3}

    const float* arow = w2 + (size_t)(wave * 16 + nl) * KC + 2 * hi;

    v8f acc0 = {0.f, 0.f, 0.f, 0.f, 0.f, 0.f, 0.f, 0.f};
    v8f acc1 = {0.f, 0.f, 0.f, 0.f, 0.f, 0.f, 0.f, 0.f};
    #pragma unroll 4
    for (int kc = 0; kc < KC; kc += 4) {
        v2f a = *(const v2f*)(arow + kc);                 // global B64, 8B aligned
        const int r0 = (kc + 2 * hi) * SPAD + nl;
        v2f b0, b1;
        b0.x = S[r0];            b0.y = S[r0 + SPAD];          // N sub-tile 0
        b1.x = S[r0 + 16];       b1.y = S[r0 + SPAD + 16];     // N sub-tile 1
        acc0 = __builtin_amdgcn_wmma_f32_16x16x4_f32(
                   false, a, false, b0, (short)0, acc0, false, false);
        acc1 = __builtin_amdgcn_wmma_f32_16x16x4_f32(
                   false, a, false, b1, (short)0, acc1, false, false);
    }

    // ---- epilogue: C/D layout: VGPR r -> M = r + 8*hi, N = nl ----
    #pragma unroll
    for (int sub = 0; sub < 2; ++sub) {
        const v8f& acc = sub ? acc1 : acc0;
        const int gp  = tile * NPIX + sub * 16 + nl;
        const int bo  = gp >> 12;
        const int ohw = gp & (HW - 1);
        #pragma unroll
        for (int r = 0; r < 8; ++r) {
            int o = wave * 16 + r + hi * 8;
            out[(size_t)(bo * OCH + o) * HW + ohw] = acc[r] + bias[o];
        }
    }
}

// ---------------------------------------------------------------------------
// Launcher.  Inputs: x, offset_w, offset_b, weight, bias (all f32).
// Workspace layout (floats): xn | offs | w2 | ow2  (~19.8 MB total).
// ---------------------------------------------------------------------------
extern "C" void kernel_launch(void* const* d_in, const int* in_sizes, int n_in,
                              void* d_out, int out_size, void* d_ws, size_t ws_size,
                              hipStream_t stream) {
    const float* x        = (const float*)d_in[0];
    const float* offset_w = (const float*)d_in[1];
    const float* offset_b = (const float*)d_in[2];
    const float* weight   = (const float*)d_in[3];
    const float* bias     = (const float*)d_in[4];
    float*       out      = (float*)d_out;

    float* ws   = (float*)d_ws;
    float* xn   = ws;                                   // 4,194,304 floats
    float* offs = xn   + (size_t)BATCH * HW * CIN;      //   589,824 floats
    float* w2   = offs + (size_t)BATCH * 18 * HW;       //   147,456 floats
    float* ow2  = w2   + (size_t)OCH * KC;              //    20,736 floats

    xpose_x_kernel    <<<(BATCH * CIN * HW) / 256, 256, 0, stream>>>(x, xn);
    xpose_w_kernel    <<<(OCH * KC + 255) / 256,   256, 0, stream>>>(weight, w2, offset_w, ow2);
    offset_conv_kernel<<<(BATCH * HW) / 256,       256, 0, stream>>>(xn, ow2, offset_b, offs);
    deform_wmma_kernel<<<(BATCH * HW) / NPIX,      256, 0, stream>>>(xn, offs, w2, bias, out);
}